// GATSkip_41334765257095
// MI455X (gfx1250) — compile-verified
//
#include <hip/hip_runtime.h>
#include <math.h>

#define NN 50000
#define NE 800000
#define C 64
// NN % 16 == 0: every live GEMM wave owns a full 16-row tile (no row guards).

typedef __attribute__((ext_vector_type(2))) float v2f;
typedef __attribute__((ext_vector_type(8))) float v8f;

// ---------------------------------------------------------------------------
// float atomic max via int/uint ordering trick (init to -inf)
// ---------------------------------------------------------------------------
__device__ __forceinline__ void atomicMaxFloat(float* addr, float v) {
    if (v >= 0.0f)
        atomicMax((int*)addr, __float_as_int(v));
    else
        atomicMin((unsigned int*)addr, __float_as_uint(v));
}

// ---------------------------------------------------------------------------
// H[n,64] (+)= A[n,64] @ W^T      (W row c at W[c*w_rowstride], k = 0..63)
// One wave computes a 16-row x 64-col stripe using V_WMMA_F32_16X16X4_F32.
// Block = 128 threads = 4 waves = 64 rows. Requires n % 16 == 0.
// ---------------------------------------------------------------------------
__global__ void gemm_wmma_f32(const float* __restrict__ A,
                              const float* __restrict__ W, int w_rowstride,
                              float* __restrict__ H, int n, int accumulate) {
    const int wave = threadIdx.x >> 5;
    const int lane = threadIdx.x & 31;
    const int row0 = (blockIdx.x * 4 + wave) * 16;
    if (row0 >= n) return;                 // wave-uniform

    const int mlo = lane & 15;             // col-in-tile for B/C, row for A frag
    const int hi  = lane >> 4;             // lane half

    v8f acc[4];
    if (accumulate) {
        for (int t = 0; t < 4; ++t)
            for (int r = 0; r < 8; ++r)
                acc[t][r] = H[(row0 + r + hi * 8) * C + t * 16 + mlo];
    } else {
        for (int t = 0; t < 4; ++t) acc[t] = (v8f){0,0,0,0,0,0,0,0};
    }

    const int arow = row0 + mlo;

    for (int kk = 0; kk < C; kk += 4) {
        const int kb = kk + hi * 2;
        // A fragment (16x4): lanes 0-15 hold K=kk,kk+1; lanes 16-31 K=kk+2,kk+3
        v2f a;
        a.x = A[arow * C + kb];
        a.y = A[arow * C + kb + 1];
        for (int t = 0; t < 4; ++t) {
            // B fragment (4x16): B[k][ncol] = W[ncol][k]
            const int bn = t * 16 + mlo;
            v2f b;
            b.x = W[bn * w_rowstride + kb];
            b.y = W[bn * w_rowstride + kb + 1];
            acc[t] = __builtin_amdgcn_wmma_f32_16x16x4_f32(
                false, a, false, b, (short)0, acc[t], false, false);
        }
    }

    for (int t = 0; t < 4; ++t)
        for (int r = 0; r < 8; ++r)
            H[(row0 + r + hi * 8) * C + t * 16 + mlo] = acc[t][r];
}

// ---------------------------------------------------------------------------
// per-node attention scores: s_src[i] = h[i,:].a_src ; s_dst[i] = h[i,:].a_dst
// one wave per node (coalesced 64-wide row reads)
// ---------------------------------------------------------------------------
__global__ void scores_kernel(const float* __restrict__ H,
                              const float* __restrict__ a_src,
                              const float* __restrict__ a_dst,
                              float* __restrict__ s_src, float* __restrict__ s_dst,
                              int n) {
    const int wave = threadIdx.x >> 5;
    const int lane = threadIdx.x & 31;
    const int i = blockIdx.x * 4 + wave;
    if (i >= n) return;
    float h0 = H[i * C + lane];
    float h1 = H[i * C + 32 + lane];
    float ps = h0 * a_src[lane] + h1 * a_src[32 + lane];
    float pd = h0 * a_dst[lane] + h1 * a_dst[32 + lane];
    for (int off = 16; off > 0; off >>= 1) {
        ps += __shfl_down(ps, off, 32);
        pd += __shfl_down(pd, off, 32);
    }
    if (lane == 0) { s_src[i] = ps; s_dst[i] = pd; }
}

// ---------------------------------------------------------------------------
// edge-attr mean per destination (layer-invariant)
// ---------------------------------------------------------------------------
__global__ void ea_accum_kernel(const int* __restrict__ dst,
                                const float* __restrict__ ea,
                                float* __restrict__ sumea, float* __restrict__ cnt) {
    int e = blockIdx.x * blockDim.x + threadIdx.x;
    if (e >= NE) return;
    int d = dst[e];
    atomicAdd(&sumea[d * 2 + 0], ea[e * 2 + 0]);
    atomicAdd(&sumea[d * 2 + 1], ea[e * 2 + 1]);
    atomicAdd(&cnt[d], 1.0f);
}

__global__ void ea_mean_kernel(float* __restrict__ sumea,
                               const float* __restrict__ cnt, int n) {
    int i = blockIdx.x * blockDim.x + threadIdx.x;
    if (i >= n) return;
    float c = fmaxf(cnt[i], 1.0f);
    sumea[i * 2 + 0] /= c;
    sumea[i * 2 + 1] /= c;
}

// q[l*2+j] = sum_c We[l][c][j] * ae[l][c]   (collapses (ea@We^T)@a_e to 2 scalars)
__global__ void qvec_kernel(const float* __restrict__ We,
                            const float* __restrict__ ae, float* __restrict__ q) {
    int t = threadIdx.x;
    if (t < 6) {
        int l = t >> 1, j = t & 1;
        float s = 0.0f;
        for (int c = 0; c < C; ++c) s += We[l * C * 2 + c * 2 + j] * ae[l * C + c];
        q[t] = s;
    } else if (t < 8) {
        q[t] = 0.0f;   // skip layer: no edge_attr
    }
}

// ---------------------------------------------------------------------------
// per-layer init: agg = 0 ; amax = -inf ; esum = 0
// ---------------------------------------------------------------------------
__global__ void layer_init_kernel(float* __restrict__ agg, float* __restrict__ amax,
                                  float* __restrict__ esum, int n) {
    int i = blockIdx.x * blockDim.x + threadIdx.x;
    if (i < n * C) agg[i] = 0.0f;
    if (i < n) { amax[i] = -__builtin_inff(); esum[i] = 0.0f; }
}

// ---------------------------------------------------------------------------
// edge alpha + leaky relu + running max per destination
// ---------------------------------------------------------------------------
__global__ void edge_alpha_kernel(const int* __restrict__ src, const int* __restrict__ dst,
                                  const float* __restrict__ ea,
                                  const float* __restrict__ ssrc, const float* __restrict__ sdst,
                                  const float* __restrict__ q2,
                                  float* __restrict__ alpha, float* __restrict__ amax) {
    int e = blockIdx.x * blockDim.x + threadIdx.x;
    if (e >= NE) return;
    int s = src[e], d = dst[e];
    float a = ssrc[s] + sdst[d] + q2[0] * ea[e * 2 + 0] + q2[1] * ea[e * 2 + 1];
    a = a > 0.0f ? a : 0.2f * a;
    alpha[e] = a;
    atomicMaxFloat(&amax[d], a);
}

// self-loop alpha (uses mean edge attr); plain max (one writer per node)
__global__ void self_alpha_kernel(const float* __restrict__ ssrc, const float* __restrict__ sdst,
                                  const float* __restrict__ meanea, const float* __restrict__ q2,
                                  float* __restrict__ aself, float* __restrict__ amax, int n) {
    int i = blockIdx.x * blockDim.x + threadIdx.x;
    if (i >= n) return;
    float a = ssrc[i] + sdst[i] + q2[0] * meanea[i * 2 + 0] + q2[1] * meanea[i * 2 + 1];
    a = a > 0.0f ? a : 0.2f * a;
    aself[i] = a;
    amax[i] = fmaxf(amax[i], a);
}

__global__ void edge_exp_kernel(const int* __restrict__ dst, float* __restrict__ alpha,
                                const float* __restrict__ amax, float* __restrict__ esum) {
    int e = blockIdx.x * blockDim.x + threadIdx.x;
    if (e >= NE) return;
    int d = dst[e];
    float w = expf(alpha[e] - amax[d]);
    alpha[e] = w;
    atomicAdd(&esum[d], w);
}

__global__ void self_exp_kernel(float* __restrict__ aself, const float* __restrict__ amax,
                                float* __restrict__ esum, int n) {
    int i = blockIdx.x * blockDim.x + threadIdx.x;
    if (i >= n) return;
    float w = expf(aself[i] - amax[i]);
    aself[i] = w;
    esum[i] += w;        // edge pass already finished (same stream ordering)
}

// ---------------------------------------------------------------------------
// scatter-add aggregation: 64 threads per edge, channel = fastest index
// h & agg (12.8 MB each) stay L2-resident on the 192 MB L2.
// ---------------------------------------------------------------------------
__global__ void edge_aggregate_kernel(const int* __restrict__ src, const int* __restrict__ dst,
                                      const float* __restrict__ alpha, const float* __restrict__ esum,
                                      const float* __restrict__ H, float* __restrict__ agg) {
    int gid = blockIdx.x * blockDim.x + threadIdx.x;
    int e = gid >> 6;
    int c = gid & 63;
    if (e >= NE) return;
    int s = src[e], d = dst[e];
    float w = alpha[e] / esum[d];
    atomicAdd(&agg[d * C + c], w * H[s * C + c]);
}

// out = agg + (aself/esum)*h + bias ; optional exact GELU
__global__ void finalize_kernel(const float* __restrict__ agg, const float* __restrict__ H,
                                const float* __restrict__ aself, const float* __restrict__ esum,
                                const float* __restrict__ bias, float* __restrict__ out,
                                int n, int do_gelu) {
    int gid = blockIdx.x * blockDim.x + threadIdx.x;
    int i = gid >> 6;
    int c = gid & 63;
    if (i >= n) return;
    float wself = aself[i] / esum[i];
    float v = agg[i * C + c] + wself * H[i * C + c] + bias[c];
    if (do_gelu) v = 0.5f * v * (1.0f + erff(v * 0.70710678118654752f));
    out[i * C + c] = v;
}

// ---------------------------------------------------------------------------
extern "C" void kernel_launch(void* const* d_in, const int* in_sizes, int n_in,
                              void* d_out, int out_size, void* d_ws, size_t ws_size,
                              hipStream_t stream) {
    const float* x          = (const float*)d_in[0];
    const int*   ei         = (const int*)  d_in[1];
    const float* ea         = (const float*)d_in[2];
    const float* W_std      = (const float*)d_in[3];
    const float* a_src_std  = (const float*)d_in[4];
    const float* a_dst_std  = (const float*)d_in[5];
    const float* We_std     = (const float*)d_in[6];
    const float* ae_std     = (const float*)d_in[7];
    const float* b_std      = (const float*)d_in[8];
    const float* W_skip     = (const float*)d_in[9];
    const float* a_src_skip = (const float*)d_in[10];
    const float* a_dst_skip = (const float*)d_in[11];
    const float* b_skip     = (const float*)d_in[12];

    const int* src = ei;
    const int* dst = ei + NE;

    // workspace layout (floats)
    float* ws = (float*)d_ws;
    size_t off = 0;
    float* h      = ws + off; off += (size_t)NN * C;   // 3.2M
    float* act    = ws + off; off += (size_t)NN * C;   // 3.2M
    float* agg    = ws + off; off += (size_t)NN * C;   // 3.2M
    float* alpha  = ws + off; off += (size_t)NE;       // 0.8M
    float* ssrc   = ws + off; off += NN;
    float* sdst   = ws + off; off += NN;
    float* amax   = ws + off; off += NN;
    float* esum   = ws + off; off += NN;
    float* aself  = ws + off; off += NN;
    float* cnt    = ws + off; off += NN;
    float* meanea = ws + off; off += 2 * (size_t)NN;
    float* q      = ws + off; off += 8;

    const int TB = 256;
    const int edge_blocks  = (NE + TB - 1) / TB;
    const int node_blocks  = (NN + TB - 1) / TB;
    const int nc_blocks    = (NN * C + TB - 1) / TB;
    const int ec_blocks    = (NE * C + TB - 1) / TB;
    const int gemm_blocks  = (NN + 63) / 64;
    const int score_blocks = (NN + 3) / 4;

    // layer-invariant pre-computation ------------------------------------
    hipMemsetAsync(meanea, 0, 2 * (size_t)NN * sizeof(float), stream);
    hipMemsetAsync(cnt,    0, (size_t)NN * sizeof(float), stream);
    ea_accum_kernel<<<edge_blocks, TB, 0, stream>>>(dst, ea, meanea, cnt);
    ea_mean_kernel<<<node_blocks, TB, 0, stream>>>(meanea, cnt, NN);
    qvec_kernel<<<1, 64, 0, stream>>>(We_std, ae_std, q);

    // 3 GAT layers with edge attributes + GELU ---------------------------
    const float* in = x;
    for (int l = 0; l < 3; ++l) {
        gemm_wmma_f32<<<gemm_blocks, 128, 0, stream>>>(in, W_std + (size_t)l * C * C, C, h, NN, 0);
        scores_kernel<<<score_blocks, 128, 0, stream>>>(h, a_src_std + l * C, a_dst_std + l * C,
                                                        ssrc, sdst, NN);
        layer_init_kernel<<<nc_blocks, TB, 0, stream>>>(agg, amax, esum, NN);
        edge_alpha_kernel<<<edge_blocks, TB, 0, stream>>>(src, dst, ea, ssrc, sdst, q + 2 * l,
                                                          alpha, amax);
        self_alpha_kernel<<<node_blocks, TB, 0, stream>>>(ssrc, sdst, meanea, q + 2 * l,
                                                          aself, amax, NN);
        edge_exp_kernel<<<edge_blocks, TB, 0, stream>>>(dst, alpha, amax, esum);
        self_exp_kernel<<<node_blocks, TB, 0, stream>>>(aself, amax, esum, NN);
        edge_aggregate_kernel<<<ec_blocks, TB, 0, stream>>>(src, dst, alpha, esum, h, agg);
        finalize_kernel<<<nc_blocks, TB, 0, stream>>>(agg, h, aself, esum, b_std + l * C,
                                                      act, NN, 1);
        in = act;
    }

    // skip layer: concat(x, act) @ W_skip^T == two accumulating GEMMs ----
    gemm_wmma_f32<<<gemm_blocks, 128, 0, stream>>>(x,   W_skip,      2 * C, h, NN, 0);
    gemm_wmma_f32<<<gemm_blocks, 128, 0, stream>>>(act, W_skip + C,  2 * C, h, NN, 1);
    scores_kernel<<<score_blocks, 128, 0, stream>>>(h, a_src_skip, a_dst_skip, ssrc, sdst, NN);
    layer_init_kernel<<<nc_blocks, TB, 0, stream>>>(agg, amax, esum, NN);
    edge_alpha_kernel<<<edge_blocks, TB, 0, stream>>>(src, dst, ea, ssrc, sdst, q + 6,
                                                      alpha, amax);     // q[6..7] == 0
    self_alpha_kernel<<<node_blocks, TB, 0, stream>>>(ssrc, sdst, meanea, q + 6,
                                                      aself, amax, NN);
    edge_exp_kernel<<<edge_blocks, TB, 0, stream>>>(dst, alpha, amax, esum);
    self_exp_kernel<<<node_blocks, TB, 0, stream>>>(aself, amax, esum, NN);
    edge_aggregate_kernel<<<ec_blocks, TB, 0, stream>>>(src, dst, alpha, esum, h, agg);
    finalize_kernel<<<nc_blocks, TB, 0, stream>>>(agg, h, aself, esum, b_skip,
                                                  (float*)d_out, NN, 0);
}